// Denoiser_57939108823686
// MI455X (gfx1250) — compile-verified
//
#include <hip/hip_runtime.h>

// Problem constants (match reference)
#define BB   64
#define MM   90
#define TT   187
#define MP   96          // M padded to 6 WMMA tiles of 16
#define KP   188         // T padded to multiple of 4 (WMMA f32 K=4)
#define NT   6           // MP/16
#define NTILES (NT*NT)   // 36
#define NWAVES 8
#define QMAX 5           // ceil(36/8) tiles per wave
#define THREADS 256
#define NNLS_ITERS 500
#define POWER_ITERS 50
#define NEG 0.33f

typedef __attribute__((ext_vector_type(2))) float v2f;
typedef __attribute__((ext_vector_type(8))) float v8f;

__device__ __forceinline__ float leaky(float x) { return x >= 0.f ? x : NEG * x; }

// D = A(16x4) * B(4x16) + C(16x16), all f32, wave32.
__device__ __forceinline__ v8f wmma_f32(v2f a, v2f b, v8f c) {
  return __builtin_amdgcn_wmma_f32_16x16x4_f32(false, a, false, b, (short)0, c, false, false);
}

__global__ __launch_bounds__(THREADS)
void denoiser_kernel(const float* __restrict__ x0_raw, const float* __restrict__ raw_x,
                     const float* __restrict__ w1g,  const float* __restrict__ b1g,
                     const float* __restrict__ w2g,  const float* __restrict__ b2g,
                     const float* __restrict__ w3g,  const float* __restrict__ b3g,
                     const float* __restrict__ w4g,  const float* __restrict__ b4g,
                     const float* __restrict__ e2nw, const float* __restrict__ e2nb,
                     const float* __restrict__ n2gw, const float* __restrict__ n2gb,
                     const float* __restrict__ d1w,  const float* __restrict__ d1b,
                     const float* __restrict__ d2w,  const float* __restrict__ d2b,
                     const float* __restrict__ d3w,  const float* __restrict__ d3b,
                     float* __restrict__ out0, float* __restrict__ out1, float* __restrict__ out2)
{
  // LDS: G(36KB) + C(36KB) + {X staging | y,w}(72KB) + scratch
  __shared__ float sG[MP * MP];       // Gram, symmetric
  __shared__ float sC[MP * MP];       // X^T b targets; reused by CNN tail
  __shared__ float sYW[2 * MP * MP];  // [y | w]; overlaid by padded X (188*96 <= 2*96*96)
  __shared__ float sScr[512];         // v, u, scalars, block reductions

  float* sY = sYW;
  float* sW = sYW + MP * MP;
  float* sX = sYW;                    // X[t][m], t<188 (padded), m<96 (padded)

  const int b    = blockIdx.x;
  const int tid  = threadIdx.x;
  const int lane = tid & 31;
  const int wave = tid >> 5;
  const int half = lane >> 4;         // 0: K lanes {0,1}, 1: K lanes {2,3}
  const int lr   = lane & 15;

  const float* xb = x0_raw + (size_t)b * MM * TT;
  const float* rb = raw_x  + (size_t)b * MM * TT;

  // ---------------- Phase A: per-batch mean / max|x| normalize ----------------
  float psum = 0.f, pmax = 0.f;
  for (int i = tid; i < MM * TT; i += THREADS) {
    float v = xb[i];
    psum += v;
    pmax = fmaxf(pmax, fabsf(v));
  }
  float* red = sScr + 256;
  red[tid] = psum; __syncthreads();
  for (int s = 128; s > 0; s >>= 1) { if (tid < s) red[tid] += red[tid + s]; __syncthreads(); }
  if (tid == 0) sScr[192] = red[0] / (float)(MM * TT);
  __syncthreads();
  red[tid] = pmax; __syncthreads();
  for (int s = 128; s > 0; s >>= 1) { if (tid < s) red[tid] = fmaxf(red[tid], red[tid + s]); __syncthreads(); }
  if (tid == 0) sScr[193] = 1.0f / red[0];
  __syncthreads();
  const float mean = sScr[192];
  const float inv  = sScr[193];

  // stage X (zero-padded), write x0_pred output
  for (int i = tid; i < 2 * MP * MP; i += THREADS) sYW[i] = 0.f;
  __syncthreads();
  for (int i = tid; i < MM * TT; i += THREADS) {
    int m = i / TT, t = i % TT;
    float v = (xb[i] - mean) * inv;
    out0[(size_t)b * MM * TT + i] = v;
    sX[t * MP + m] = v;               // transposed: X[t][m]
  }
  __syncthreads();

  // ---------------- Phase B: G = X^T X and C = X^T raw_x via f32 WMMA ----------------
  for (int tile = wave; tile < NTILES; tile += NWAVES) {
    const int ti = tile / NT, tj = tile % NT;
    v8f acc = {};
    for (int kk = 0; kk < KP / 4; ++kk) {
      const int k0 = kk * 4 + 2 * half;
      v2f a, bf;
      a.x  = sX[(k0    ) * MP + ti * 16 + lr];
      a.y  = sX[(k0 + 1) * MP + ti * 16 + lr];
      bf.x = sX[(k0    ) * MP + tj * 16 + lr];
      bf.y = sX[(k0 + 1) * MP + tj * 16 + lr];
      acc = wmma_f32(a, bf, acc);
    }
    #pragma unroll
    for (int v = 0; v < 8; ++v)
      sG[(ti * 16 + v + 8 * half) * MP + tj * 16 + lr] = acc[v];
  }
  // C[j,m] = sum_t raw_x[j,t] * X[t,m]
  for (int tile = wave; tile < NTILES; tile += NWAVES) {
    const int ti = tile / NT, tj = tile % NT;
    v8f acc = {};
    const int j = ti * 16 + lr;
    for (int kk = 0; kk < KP / 4; ++kk) {
      const int k0 = kk * 4 + 2 * half;
      v2f a, bf;
      float a0 = 0.f, a1 = 0.f;
      if (j < MM) {
        if (k0     < TT) a0 = rb[j * TT + k0];
        if (k0 + 1 < TT) a1 = rb[j * TT + k0 + 1];
      }
      a.x = a0; a.y = a1;
      bf.x = sX[(k0    ) * MP + tj * 16 + lr];
      bf.y = sX[(k0 + 1) * MP + tj * 16 + lr];
      acc = wmma_f32(a, bf, acc);
    }
    #pragma unroll
    for (int v = 0; v < 8; ++v)
      sC[(ti * 16 + v + 8 * half) * MP + tj * 16 + lr] = acc[v];
  }
  __syncthreads();

  // ---------------- Phase C: power iteration for Lipschitz constant ----------------
  float* pv = sScr;        // v[96]
  float* pu = sScr + 96;   // u[96]
  if (tid < MP) pv[tid] = (tid < MM) ? 1.0f : 0.0f;
  __syncthreads();
  for (int it = 0; it < POWER_ITERS; ++it) {
    if (tid < MP) {
      float s = 0.f;
      for (int n = 0; n < MP; ++n) s += sG[tid * MP + n] * pv[n];
      pu[tid] = s;
    }
    __syncthreads();
    if (tid == 0) {
      float ss = 0.f;
      for (int n = 0; n < MP; ++n) ss += pu[n] * pu[n];
      sScr[194] = 1.0f / (sqrtf(ss) + 1e-12f);
    }
    __syncthreads();
    if (tid < MP) pv[tid] = pu[tid] * sScr[194];
    __syncthreads();
  }
  if (tid < MP) {
    float s = 0.f;
    for (int n = 0; n < MP; ++n) s += sG[tid * MP + n] * pv[n];
    pu[tid] = s;
  }
  __syncthreads();
  if (tid == 0) {
    float l = 0.f;
    for (int n = 0; n < MP; ++n) l += pv[n] * pu[n];
    sScr[195] = 1.0f / (l * 1.01f + 1e-8f);
  }
  __syncthreads();
  const float step = sScr[195];

  // ---------------- Hoist loop-invariants into registers ----------------
  // Transposed formulation: D^T[m,j] = sum_n G[m,n] * y[j,n]  (G symmetric)
  //   A = G rows (loop-invariant -> registers), B = y rows (contiguous b64 LDS loads)
  v2f   garr[QMAX][MP / 4];   // 240 VGPRs of G fragments
  float creg[QMAX][8];        // C[j,m] per owned output element
  float wreg[QMAX][8];        // FISTA w state per owned output element
  #pragma unroll
  for (int q = 0; q < QMAX; ++q) {
    const int tile = wave + NWAVES * q;
    if (tile < NTILES) {
      const int ti = tile / NT, tj = tile % NT;
      const int rowG = (ti * 16 + lr) * MP;
      #pragma unroll
      for (int kk = 0; kk < MP / 4; ++kk) {
        const int k0 = kk * 4 + 2 * half;
        garr[q][kk].x = sG[rowG + k0];
        garr[q][kk].y = sG[rowG + k0 + 1];
      }
      const int rowC = (tj * 16 + lr) * MP + ti * 16 + 8 * half;
      #pragma unroll
      for (int v = 0; v < 8; ++v) {
        creg[q][v] = sC[rowC + v];
        wreg[q][v] = 0.f;
      }
    }
  }
  __syncthreads();

  // zero y, w for FISTA (frees the X staging alias; w kept in LDS only for final dump)
  for (int i = tid; i < 2 * MP * MP; i += THREADS) sYW[i] = 0.f;
  __syncthreads();

  // ---------------- Phase D: FISTA, 500 iterations of grad^T = G * y^T ----------------
  float tcur = 1.0f;
  for (int it = 0; it < NNLS_ITERS; ++it) {
    const float t1  = 0.5f * (1.0f + sqrtf(1.0f + 4.0f * tcur * tcur));
    const float mom = (tcur - 1.0f) / t1;
    tcur = t1;

    v8f acc[QMAX];
    #pragma unroll
    for (int q = 0; q < QMAX; ++q) {
      const int tile = wave + NWAVES * q;      // uniform per wave -> EXEC all-1s at WMMA
      if (tile < NTILES) {
        const int tj = tile % NT;
        const int rowY = (tj * 16 + lr) * MP;  // B[k,j] = y[j,k]: row of y, contiguous
        v8f a8 = {};
        #pragma unroll
        for (int kk = 0; kk < MP / 4; ++kk) {
          const int k0 = kk * 4 + 2 * half;
          v2f bf;
          bf.x = sY[rowY + k0];
          bf.y = sY[rowY + k0 + 1];
          a8 = wmma_f32(garr[q][kk], bf, a8);
        }
        acc[q] = a8;
      }
    }
    __syncthreads();   // all reads of y complete before updates
    #pragma unroll
    for (int q = 0; q < QMAX; ++q) {
      const int tile = wave + NWAVES * q;
      if (tile < NTILES) {
        const int ti = tile / NT, tj = tile % NT;
        const int jr = tj * 16 + lr;                 // output column = j index
        #pragma unroll
        for (int v = 0; v < 8; ++v) {
          const int mr = ti * 16 + v + 8 * half;     // output row = m index
          float g = acc[q][v] - creg[q][v];
          if (jr == mr) g = 0.f;                     // mask = 1 - eye
          const float yv   = sY[jr * MP + mr];
          const float wold = wreg[q][v];
          const float w1   = fmaxf(yv - step * g, 0.f);
          wreg[q][v] = w1;
          sY[jr * MP + mr] = w1 + mom * (w1 - wold);
        }
      }
    }
    __syncthreads();
  }

  // dump w registers to LDS once for output + CNN phases
  #pragma unroll
  for (int q = 0; q < QMAX; ++q) {
    const int tile = wave + NWAVES * q;
    if (tile < NTILES) {
      const int ti = tile / NT, tj = tile % NT;
      const int jr = tj * 16 + lr;
      #pragma unroll
      for (int v = 0; v < 8; ++v)
        sW[jr * MP + ti * 16 + v + 8 * half] = wreg[q][v];
    }
  }
  __syncthreads();

  // write A = w (diag/padding are zero by construction)
  for (int i = tid; i < MM * MM; i += THREADS) {
    int j = i / MM, m = i % MM;
    out1[(size_t)b * MM * MM + i] = sW[j * MP + m];
  }
  __syncthreads();

  // ---------------- Phase E: BrainNetCNN tail (tiny, VALU) ----------------
  float* sa  = sC;          // a  [8][90]
  float* sb  = sC + 720;    // b  [8][90]
  float* sa2 = sC + 1440;   // a2 [16][90]
  float* sb2 = sC + 2880;   // b2 [16][90]
  float* se  = sC + 4320;   // e2n [90]
  float* sn  = sC + 4416;   // n2g [64]
  float* sh1 = sC + 4480;   // [128]
  float* sh2 = sC + 4608;   // [10]

  for (int i = tid; i < 8 * MM; i += THREADS) {
    int o = i / MM, h = i % MM;
    float acc = b1g[o];
    for (int w = 0; w < MM; ++w) acc += sW[h * MP + w] * w1g[o * MM + w];
    sa[o * MM + h] = acc;
  }
  for (int i = tid; i < 8 * MM; i += THREADS) {
    int o = i / MM, w = i % MM;
    float acc = b2g[o];
    for (int h = 0; h < MM; ++h) acc += sW[h * MP + w] * w2g[o * MM + h];
    sb[o * MM + w] = acc;
  }
  __syncthreads();
  for (int i = tid; i < 16 * MM; i += THREADS) {
    int o = i / MM, h = i % MM;
    float acc = b3g[o];
    for (int c = 0; c < 8; ++c) {
      float ac = sa[c * MM + h];
      for (int w = 0; w < MM; ++w)
        acc += leaky(ac + sb[c * MM + w]) * w3g[(o * 8 + c) * MM + w];
    }
    sa2[o * MM + h] = acc;
  }
  for (int i = tid; i < 16 * MM; i += THREADS) {
    int o = i / MM, w = i % MM;
    float acc = b4g[o];
    for (int c = 0; c < 8; ++c) {
      float bc = sb[c * MM + w];
      for (int h = 0; h < MM; ++h)
        acc += leaky(sa[c * MM + h] + bc) * w4g[(o * 8 + c) * MM + h];
    }
    sb2[o * MM + w] = acc;
  }
  __syncthreads();
  if (tid < MM) {
    float acc = e2nb[0];
    for (int c = 0; c < 16; ++c) {
      float ac = sa2[c * MM + tid];
      for (int w = 0; w < MM; ++w)
        acc += leaky(ac + sb2[c * MM + w]) * e2nw[c * MM + w];
    }
    se[tid] = leaky(acc);
  }
  __syncthreads();
  if (tid < 64) {
    float acc = n2gb[tid];
    for (int h = 0; h < MM; ++h) acc += se[h] * n2gw[tid * MM + h];
    sn[tid] = leaky(acc);
  }
  __syncthreads();
  if (tid < 128) {
    float acc = d1b[tid];
    for (int k = 0; k < 64; ++k) acc += sn[k] * d1w[tid * 64 + k];
    sh1[tid] = leaky(acc);
  }
  __syncthreads();
  if (tid < 10) {
    float acc = d2b[tid];
    for (int k = 0; k < 128; ++k) acc += sh1[k] * d2w[tid * 128 + k];
    sh2[tid] = leaky(acc);
  }
  __syncthreads();
  if (tid < 2) {
    float acc = d3b[tid];
    for (int k = 0; k < 10; ++k) acc += sh2[k] * d3w[tid * 10 + k];
    out2[(size_t)b * 2 + tid] = leaky(acc);
  }
}

extern "C" void kernel_launch(void* const* d_in, const int* in_sizes, int n_in,
                              void* d_out, int out_size, void* d_ws, size_t ws_size,
                              hipStream_t stream) {
  const float* x0  = (const float*)d_in[0];
  const float* rx  = (const float*)d_in[1];
  const float* w1  = (const float*)d_in[2];
  const float* b1  = (const float*)d_in[3];
  const float* w2  = (const float*)d_in[4];
  const float* b2  = (const float*)d_in[5];
  const float* w3  = (const float*)d_in[6];
  const float* b3  = (const float*)d_in[7];
  const float* w4  = (const float*)d_in[8];
  const float* b4  = (const float*)d_in[9];
  const float* ew  = (const float*)d_in[10];
  const float* eb  = (const float*)d_in[11];
  const float* nw  = (const float*)d_in[12];
  const float* nb  = (const float*)d_in[13];
  const float* d1w = (const float*)d_in[14];
  const float* d1b = (const float*)d_in[15];
  const float* d2w = (const float*)d_in[16];
  const float* d2b = (const float*)d_in[17];
  const float* d3w = (const float*)d_in[18];
  const float* d3b = (const float*)d_in[19];

  float* out  = (float*)d_out;
  float* out0 = out;                                  // x0_pred [B,1,M,T]
  float* out1 = out0 + (size_t)BB * MM * TT;          // A [B,M,M]
  float* out2 = out1 + (size_t)BB * MM * MM;          // cls [B,2]

  denoiser_kernel<<<dim3(BB), dim3(THREADS), 0, stream>>>(
      x0, rx, w1, b1, w2, b2, w3, b3, w4, b4,
      ew, eb, nw, nb, d1w, d1b, d2w, d2b, d3w, d3b,
      out0, out1, out2);
}